// SimpleGCNLayer_60043642798088
// MI455X (gfx1250) — compile-verified
//
#include <hip/hip_runtime.h>

// Problem constants (match reference)
#define NUM_NODES 100000
#define DIM       64
#define NUM_EDGES 1250000
#define ROW_BLOCKS (NUM_NODES / 16)   // 6250, exact

typedef __attribute__((ext_vector_type(2))) float v2f;
typedef __attribute__((ext_vector_type(8))) float v8f;

// ---------------------------------------------------------------------------
// Kernel 1: zero the workspace (neigh accumulator + degree counters)
// ---------------------------------------------------------------------------
__global__ void gcn_zero_kernel(float* __restrict__ ws, int n) {
    int i = blockIdx.x * blockDim.x + threadIdx.x;
    if (i < n) ws[i] = 0.0f;
}

// ---------------------------------------------------------------------------
// Kernel 2: degree accumulation, one thread per edge
// ---------------------------------------------------------------------------
__global__ void gcn_deg_kernel(const int* __restrict__ edge_index,
                               float* __restrict__ deg) {
    int e = blockIdx.x * blockDim.x + threadIdx.x;
    if (e < NUM_EDGES) {
        int d = edge_index[NUM_EDGES + e];   // dst row of edge_index
        atomicAdd(&deg[d], 1.0f);
    }
}

// ---------------------------------------------------------------------------
// Kernel 3: feature scatter-add. One thread per (edge, feature) -> fully
// coalesced gather of x[src] rows; atomics land in L2 (agg is 25.6MB << 192MB)
// ---------------------------------------------------------------------------
__global__ void gcn_scatter_kernel(const float* __restrict__ x,
                                   const int*   __restrict__ edge_index,
                                   float*       __restrict__ agg) {
    unsigned int i = blockIdx.x * blockDim.x + threadIdx.x;
    if (i < (unsigned int)NUM_EDGES * DIM) {
        int e = (int)(i >> 6);
        int f = (int)(i & 63);
        int s = edge_index[e];                 // src
        int d = edge_index[NUM_EDGES + e];     // dst
        atomicAdd(&agg[d * DIM + f], x[s * DIM + f]);
    }
}

// ---------------------------------------------------------------------------
// Kernel 4: fused dual-GEMM + bias + ReLU via V_WMMA_F32_16X16X4_F32.
// One wave32 per 16-row block; 4 col-tiles of 16 (OUT_DIM=64) kept in 4
// f32 16x16 accumulators; K=64 walked in steps of 4; two WMMAs per step
// (self path and mean-neighbor path) accumulate into the same C.
// ---------------------------------------------------------------------------
__global__ __launch_bounds__(256)
void gcn_wmma_kernel(const float* __restrict__ x,
                     const float* __restrict__ agg,
                     const float* __restrict__ deg,
                     const float* __restrict__ W_self,   // [64][64] row-major
                     const float* __restrict__ b_self,   // [64]
                     const float* __restrict__ W_neigh,  // [64][64] row-major
                     float*       __restrict__ out) {
    const int lane = threadIdx.x & 31;
    const int wave = blockIdx.x * (blockDim.x >> 5) + (threadIdx.x >> 5);
    if (wave >= ROW_BLOCKS) return;           // uniform per-wave exit: EXEC all-1 inside

    const int row0 = wave * 16;
    const int half = lane >> 4;               // 0: K pair {0,1}, 1: K pair {2,3}
    const int lm   = lane & 15;               // A row / B col / C col within tile

    const float* xrow = x   + (size_t)(row0 + lm) * DIM;
    const float* arow = agg + (size_t)(row0 + lm) * DIM;
    const float  invd = 1.0f / fmaxf(deg[row0 + lm], 1.0f);

    v8f c0 = {}, c1 = {}, c2 = {}, c3 = {};

    #pragma unroll
    for (int kk = 0; kk < DIM; kk += 4) {
        const int k = kk + half * 2;
        // A fragments (16x4 f32): lane holds row lm, two consecutive K values
        v2f a_s = { xrow[k], xrow[k + 1] };
        v2f a_n = { arow[k] * invd, arow[k + 1] * invd };

        #pragma unroll
        for (int n = 0; n < 4; ++n) {
            // B fragments (4x16 f32): B[k][col] = W[col][k] -> contiguous pair in W row
            const float* ws_r = W_self  + (size_t)(n * 16 + lm) * DIM;
            const float* wn_r = W_neigh + (size_t)(n * 16 + lm) * DIM;
            v2f b_s = { ws_r[k], ws_r[k + 1] };
            v2f b_n = { wn_r[k], wn_r[k + 1] };

            v8f& c = (n == 0) ? c0 : (n == 1) ? c1 : (n == 2) ? c2 : c3;
            c = __builtin_amdgcn_wmma_f32_16x16x4_f32(
                    false, a_s, false, b_s, (short)0, c, false, false);
            c = __builtin_amdgcn_wmma_f32_16x16x4_f32(
                    false, a_n, false, b_n, (short)0, c, false, false);
        }
    }

    // Epilogue: bias + ReLU + store.
    // C layout: VGPR v, lane -> row = v + half*8, col = lm (within tile)
    #pragma unroll
    for (int n = 0; n < 4; ++n) {
        const v8f& c = (n == 0) ? c0 : (n == 1) ? c1 : (n == 2) ? c2 : c3;
        const int col  = n * 16 + lm;
        const float bb = b_self[col];
        #pragma unroll
        for (int v = 0; v < 8; ++v) {
            const int row = row0 + v + half * 8;
            out[(size_t)row * DIM + col] = fmaxf(c[v] + bb, 0.0f);
        }
    }
}

// ---------------------------------------------------------------------------
// Launcher
// ---------------------------------------------------------------------------
extern "C" void kernel_launch(void* const* d_in, const int* in_sizes, int n_in,
                              void* d_out, int out_size, void* d_ws, size_t ws_size,
                              hipStream_t stream) {
    const float* x          = (const float*)d_in[0];
    const int*   edge_index = (const int*)  d_in[1];
    const float* W_self     = (const float*)d_in[2];
    const float* b_self     = (const float*)d_in[3];
    const float* W_neigh    = (const float*)d_in[4];
    float*       out        = (float*)d_out;

    float* agg = (float*)d_ws;                 // NUM_NODES * DIM floats
    float* deg = agg + (size_t)NUM_NODES * DIM; // NUM_NODES floats

    // 1) zero accumulators (every call -> deterministic w.r.t. workspace state)
    {
        int n = NUM_NODES * DIM + NUM_NODES;
        int blk = 256, grd = (n + blk - 1) / blk;
        gcn_zero_kernel<<<grd, blk, 0, stream>>>((float*)d_ws, n);
    }
    // 2) degrees
    {
        int blk = 256, grd = (NUM_EDGES + blk - 1) / blk;
        gcn_deg_kernel<<<grd, blk, 0, stream>>>(edge_index, deg);
    }
    // 3) feature scatter-add
    {
        unsigned int total = (unsigned int)NUM_EDGES * DIM;
        int blk = 256;
        unsigned int grd = (total + blk - 1) / blk;
        gcn_scatter_kernel<<<grd, blk, 0, stream>>>(x, edge_index, agg);
    }
    // 4) fused WMMA GEMMs + bias + ReLU
    {
        int blk = 256;                         // 8 waves per block
        int wavesPerBlock = blk / 32;
        int grd = (ROW_BLOCKS + wavesPerBlock - 1) / wavesPerBlock;
        gcn_wmma_kernel<<<grd, blk, 0, stream>>>(x, agg, deg,
                                                 W_self, b_self, W_neigh, out);
    }
}